// DRSolverQP_26792005992478
// MI455X (gfx1250) — compile-verified
//
#include <hip/hip_runtime.h>
#include <math.h>

// Problem constants (match reference)
#define X_DIM    128
#define N_EQ     64
#define PARM_DIM 192
#define BATCH    1024
#define KDIM     192            // X_DIM + N_EQ
#define LDK      193            // padded LDS row stride (odd -> conflict-free columns)
#define NB       16             // LU block size (WMMA tile)
#define NBLK     12             // KDIM / NB
#define NTHREADS 256
#define NWAVES   8
#define GAMMA_F  1.0f
#define NUM_STEPS 3

// log bounds for the two pdiag heads
#define LLB1 (-18.420680743952367f)   // ln(1e-8)
#define LUB1 ( 18.420680743952367f)   // ln(1e8)
#define LLB2 (-0.6931471805599453f)   // ln(0.5)
#define LUB2 ( 0.6931471805599453f)   // ln(2.0)

typedef float v2f __attribute__((ext_vector_type(2)));
typedef float v8f __attribute__((ext_vector_type(8)));

// ---------------------------------------------------------------------------
// Build K = [[2Q + Pd*I, A^T],[A, 0]] into LDS (192 x 192, stride LDK)
// ---------------------------------------------------------------------------
__device__ __forceinline__ void build_K(float* __restrict__ K,
                                        const float* __restrict__ Q,
                                        const float* __restrict__ A,
                                        const float* __restrict__ Pd, int tid) {
  for (int idx = tid; idx < KDIM * KDIM; idx += NTHREADS) {
    const int i = idx / KDIM;
    const int j = idx - i * KDIM;
    float v;
    if (i < X_DIM) {
      if (j < X_DIM) {
        v = 2.0f * Q[i * X_DIM + j];
        if (i == j) v += GAMMA_F * Pd[i];
      } else {
        v = A[(j - X_DIM) * X_DIM + i];          // A^T
      }
    } else {
      v = (j < X_DIM) ? A[(i - X_DIM) * X_DIM + j] : 0.0f;
    }
    K[i * LDK + j] = v;
  }
}

// ---------------------------------------------------------------------------
// Blocked right-looking LU (no pivoting; valid for KKT with SPD H block).
// Trailing Schur updates use V_WMMA_F32_16X16X4_F32.
// ---------------------------------------------------------------------------
__device__ __forceinline__ void lu_factor_192(float* __restrict__ K,
                                              int tid, int wave, int lane) {
  const int lr16 = (lane < NB) ? lane : 0;   // clamped lane->row map for wave0 work
  const int lrow = lane & 15;
  const int lhi  = lane >> 4;

  for (int kb = 0; kb < NBLK; ++kb) {
    const int kb16 = kb * NB;

    // ---- 1) 16x16 diagonal LU: wave 0, lane r owns row r, shfl broadcasts ----
    if (wave == 0) {
      float d[NB];
      #pragma unroll
      for (int c = 0; c < NB; ++c)
        d[c] = K[(kb16 + lr16) * LDK + (kb16 + c)];
      #pragma unroll
      for (int j = 0; j < NB; ++j) {
        const float piv = __shfl(d[j], j, 32);
        const float lij = d[j] / piv;
        #pragma unroll
        for (int c = j + 1; c < NB; ++c) {
          const float ujc = __shfl(d[c], j, 32);
          if (lane > j && lane < NB) d[c] -= lij * ujc;
        }
        if (lane > j && lane < NB) d[j] = lij;
      }
      if (lane < NB) {
        #pragma unroll
        for (int c = 0; c < NB; ++c)
          K[(kb16 + lane) * LDK + (kb16 + c)] = d[c];
      }
    }
    __syncthreads();

    const int nrem = NBLK - 1 - kb;
    if (nrem == 0) break;

    // ---- 2) panels: row panel (L^-1 * B, one thread per column) and
    //               column panel (B * U^-1, one thread per row) ----
    for (int t = tid; t < nrem * 2 * NB; t += NTHREADS) {
      if (t < nrem * NB) {
        const int cc = kb16 + NB + t;                       // global column
        float y[NB];
        #pragma unroll
        for (int rr = 0; rr < NB; ++rr) {
          float v = K[(kb16 + rr) * LDK + cc];
          #pragma unroll
          for (int s = 0; s < rr; ++s)
            v -= K[(kb16 + rr) * LDK + (kb16 + s)] * y[s];  // unit-lower L
          y[rr] = v;
        }
        #pragma unroll
        for (int rr = 0; rr < NB; ++rr)
          K[(kb16 + rr) * LDK + cc] = y[rr];
      } else {
        const int R = kb16 + NB + (t - nrem * NB);          // global row
        float xr[NB];
        #pragma unroll
        for (int c = 0; c < NB; ++c) {
          float v = K[R * LDK + (kb16 + c)];
          #pragma unroll
          for (int s = 0; s < c; ++s)
            v -= xr[s] * K[(kb16 + s) * LDK + (kb16 + c)];
          xr[c] = v / K[(kb16 + c) * LDK + (kb16 + c)];
        }
        #pragma unroll
        for (int c = 0; c < NB; ++c)
          K[R * LDK + (kb16 + c)] = xr[c];
      }
    }
    __syncthreads();

    // ---- 3) trailing update C -= L_panel * U_panel via WMMA f32 16x16x4 ----
    // A layout : lane<16 -> (K=0,K=1) of row M=lane ; lane>=16 -> (K=2,K=3) of M=lane-16
    // B layout : lane<16 -> rows K=0,K=1 col N=lane ; lane>=16 -> rows K=2,K=3 col lane-16
    // C layout : VGPR r -> (M=r+8*hi, N=lane&15)
    const int ntiles = nrem * nrem;
    for (int t = wave; t < ntiles; t += NWAVES) {           // wave-uniform: EXEC full
      const int ci = (kb + 1 + t / nrem) * NB;
      const int cj = (kb + 1 + t % nrem) * NB;
      v8f acc;
      #pragma unroll
      for (int r = 0; r < 8; ++r)
        acc[r] = K[(ci + lhi * 8 + r) * LDK + (cj + lrow)];
      #pragma unroll
      for (int k4 = 0; k4 < 4; ++k4) {
        const int kc = kb16 + k4 * 4;
        v2f a, b;
        a.x = -K[(ci + lrow) * LDK + (kc + lhi * 2 + 0)];   // negate A => C - A*B
        a.y = -K[(ci + lrow) * LDK + (kc + lhi * 2 + 1)];
        b.x =  K[(kc + lhi * 2 + 0) * LDK + (cj + lrow)];
        b.y =  K[(kc + lhi * 2 + 1) * LDK + (cj + lrow)];
        acc = __builtin_amdgcn_wmma_f32_16x16x4_f32(false, a, false, b,
                                                    (short)0, acc, false, false);
      }
      #pragma unroll
      for (int r = 0; r < 8; ++r)
        K[(ci + lhi * 8 + r) * LDK + (cj + lrow)] = acc[r];
    }
    __syncthreads();
  }
}

// ---------------------------------------------------------------------------
// Solve L U x = rhs with the in-place factor (L unit-lower, U upper).
// ---------------------------------------------------------------------------
__device__ __forceinline__ void lu_solve_192(const float* __restrict__ K,
                                             float* __restrict__ rhs,
                                             int tid, int wave, int lane) {
  const int lr16 = (lane < NB) ? lane : 0;

  // forward substitution (unit lower)
  for (int kb = 0; kb < NBLK; ++kb) {
    const int kb16 = kb * NB;
    if (wave == 0) {
      float x = rhs[kb16 + lr16];
      float lr[NB];
      #pragma unroll
      for (int c = 0; c < NB; ++c)
        lr[c] = K[(kb16 + lr16) * LDK + (kb16 + c)];
      #pragma unroll
      for (int s = 0; s < NB - 1; ++s) {
        const float xs = __shfl(x, s, 32);
        if (lane > s && lane < NB) x -= lr[s] * xs;
      }
      if (lane < NB) rhs[kb16 + lane] = x;
    }
    __syncthreads();
    for (int i = kb16 + NB + tid; i < KDIM; i += NTHREADS) {
      float v = rhs[i];
      #pragma unroll
      for (int s = 0; s < NB; ++s)
        v -= K[i * LDK + (kb16 + s)] * rhs[kb16 + s];
      rhs[i] = v;
    }
    __syncthreads();
  }

  // backward substitution (upper, with diagonal)
  for (int kb = NBLK - 1; kb >= 0; --kb) {
    const int kb16 = kb * NB;
    if (wave == 0) {
      float x = rhs[kb16 + lr16];
      float ur[NB];
      #pragma unroll
      for (int c = 0; c < NB; ++c)
        ur[c] = K[(kb16 + lr16) * LDK + (kb16 + c)];
      #pragma unroll
      for (int s = NB - 1; s >= 0; --s) {
        if (lane == s) x /= ur[s];
        const float xs = __shfl(x, s, 32);
        if (lane < s) x -= ur[s] * xs;
      }
      if (lane < NB) rhs[kb16 + lane] = x;
    }
    __syncthreads();
    for (int i = tid; i < kb16; i += NTHREADS) {
      float v = rhs[i];
      #pragma unroll
      for (int s = 0; s < NB; ++s)
        v -= K[i * LDK + (kb16 + s)] * rhs[kb16 + s];
      rhs[i] = v;
    }
    __syncthreads();
  }
}

// ---------------------------------------------------------------------------
// One workgroup per batch element: pdiag heads + 2 factorizations + 4 solves.
// ---------------------------------------------------------------------------
__global__ __launch_bounds__(NTHREADS)
void drsolver_qp_kernel(const float* __restrict__ x_in,
                        const float* __restrict__ parms,
                        const float* __restrict__ Q,
                        const float* __restrict__ A,
                        const float* __restrict__ Wp1,
                        const float* __restrict__ bp1,
                        const float* __restrict__ Wp2,
                        const float* __restrict__ bp2,
                        float* __restrict__ out_x,
                        float* __restrict__ out_gap) {
  __shared__ float Ksh[KDIM * LDK];        // ~145 KB
  __shared__ float psh[PARM_DIM];
  __shared__ float Pd1[X_DIM], Pd2[X_DIM];
  __shared__ float xv[X_DIM], ysh[X_DIM], gap[X_DIM];
  __shared__ float rhs[KDIM];

  const int b    = blockIdx.x;
  const int tid  = threadIdx.x;
  const int wave = tid >> 5;
  const int lane = tid & 31;

  if (tid < PARM_DIM) psh[tid] = parms[b * PARM_DIM + tid];
  if (tid < X_DIM)    xv[tid]  = x_in[b * X_DIM + tid];
  __syncthreads();

  // ---- pdiag heads: thread t<128 -> Pd1 row t ; t>=128 -> Pd2 row t-128 ----
  {
    const int row        = tid & (X_DIM - 1);
    const float* W       = (tid < X_DIM) ? Wp1 : Wp2;
    const float* bp      = (tid < X_DIM) ? bp1 : bp2;
    const float llb      = (tid < X_DIM) ? LLB1 : LLB2;
    const float lub      = (tid < X_DIM) ? LUB1 : LUB2;
    float acc = 0.0f;
    for (int k = 0; k < PARM_DIM; ++k)
      acc = fmaf(psh[k], W[row * PARM_DIM + k], acc);
    acc += bp[row];
    const float s  = 1.0f / (1.0f + expf(-acc));
    const float pd = expf(llb + (lub - llb) * s);
    if (tid < X_DIM) Pd1[row] = pd; else Pd2[row] = pd;
  }
  __syncthreads();

  // ================= phase 1: one DR step with Pd1 =================
  build_K(Ksh, Q, A, Pd1, tid);
  if (tid < X_DIM) {
    const float xk = xv[tid];
    const float yk = fmaxf(xk, 0.0f);              // relu
    ysh[tid] = yk;
    const float v = 2.0f * yk - xk;
    rhs[tid] = GAMMA_F * Pd1[tid] * v - psh[tid];  // p = parms[:,:128]
  } else if (tid < KDIM) {
    rhs[tid] = psh[tid];                           // beq = parms[:,128:192]
  }
  __syncthreads();
  lu_factor_192(Ksh, tid, wave, lane);
  lu_solve_192(Ksh, rhs, tid, wave, lane);
  if (tid < X_DIM) xv[tid] += rhs[tid] - ysh[tid]; // x += z - y
  __syncthreads();

  // ============ phase 2: factor once with Pd2, 3 cheap solves ============
  build_K(Ksh, Q, A, Pd2, tid);
  __syncthreads();
  lu_factor_192(Ksh, tid, wave, lane);

  for (int step = 0; step < NUM_STEPS; ++step) {
    if (tid < X_DIM)      rhs[tid] = GAMMA_F * Pd2[tid] * xv[tid] - psh[tid];
    else if (tid < KDIM)  rhs[tid] = psh[tid];
    __syncthreads();
    lu_solve_192(Ksh, rhs, tid, wave, lane);
    if (tid < X_DIM) {
      const float yk = rhs[tid];                   // y = qp_prox(x)
      const float xk = xv[tid];
      const float zk = fmaxf(2.0f * yk - xk, 0.0f);
      gap[tid] = zk - yk;
      xv[tid]  = xk + (zk - yk);
    }
    __syncthreads();
  }

  if (tid < X_DIM) {
    out_x[b * X_DIM + tid]   = xv[tid];
    out_gap[b * X_DIM + tid] = gap[tid];
  }
}

// ---------------------------------------------------------------------------
extern "C" void kernel_launch(void* const* d_in, const int* in_sizes, int n_in,
                              void* d_out, int out_size, void* d_ws, size_t ws_size,
                              hipStream_t stream) {
  (void)in_sizes; (void)n_in; (void)out_size; (void)d_ws; (void)ws_size;
  const float* x     = (const float*)d_in[0];
  const float* parms = (const float*)d_in[1];
  const float* Q     = (const float*)d_in[2];
  const float* A     = (const float*)d_in[3];
  const float* Wp1   = (const float*)d_in[4];
  const float* bp1   = (const float*)d_in[5];
  const float* Wp2   = (const float*)d_in[6];
  const float* bp2   = (const float*)d_in[7];
  float* out = (float*)d_out;
  drsolver_qp_kernel<<<BATCH, NTHREADS, 0, stream>>>(
      x, parms, Q, A, Wp1, bp1, Wp2, bp2, out, out + (size_t)BATCH * X_DIM);
}